// Sp_GCN_89902255440937
// MI455X (gfx1250) — compile-verified
//
#include <hip/hip_runtime.h>

#define D 128
#define MTILES 4   // row-tiles (of 16) per block in the GEMM

typedef __attribute__((ext_vector_type(2))) float v2f;
typedef __attribute__((ext_vector_type(8))) float v8f;

// ---------------------------------------------------------------------------
// Zero a buffer (float4 granularity). Needed because scatter accumulates with
// atomics and the harness does not re-zero d_out/d_ws between replays.
// ---------------------------------------------------------------------------
__global__ __launch_bounds__(256) void gcn_zero_kernel(float4* __restrict__ p, long n4) {
    long i = (long)blockIdx.x * blockDim.x + threadIdx.x;
    if (i < n4) p[i] = make_float4(0.f, 0.f, 0.f, 0.f);
}

// ---------------------------------------------------------------------------
// Final elementwise ReLU on the output buffer.
// ---------------------------------------------------------------------------
__global__ __launch_bounds__(256) void gcn_relu_kernel(float4* __restrict__ p, long n4) {
    long i = (long)blockIdx.x * blockDim.x + threadIdx.x;
    if (i < n4) {
        float4 x = p[i];
        x.x = fmaxf(x.x, 0.f);
        x.y = fmaxf(x.y, 0.f);
        x.z = fmaxf(x.z, 0.f);
        x.w = fmaxf(x.w, 0.f);
        p[i] = x;
    }
}

// ---------------------------------------------------------------------------
// Dense GEMM  out[nrows x 128] = act(A)[nrows x 128] @ W[128 x 128]
// Full fp32 via V_WMMA_F32_16X16X4_F32 (keeps reference numerics; GEMM is a
// tiny share of runtime so fp32 matrix path is the right precision choice).
//
// Block = 256 threads = 8 waves. Wave w owns output columns [16w, 16w+16).
// W is staged once into LDS (64 KB; WGP has 320 KB) and reused across MTILES
// row-tiles. All branching around WMMA is uniform or handled by clamping, so
// EXEC is all-ones at every WMMA (ISA requirement).
//
// fp32 WMMA VGPR layouts (ISA 7.12.2):
//   A 16x4 : lane l(half h): VGPR pair = A[m0+l][k0+2h], A[m0+l][k0+2h+1]
//   B 4x16 : lane l(half h): VGPR pair = W[k0+2h][n0+l], W[k0+2h+1][n0+l]
//   C 16x16: VGPR j, lane l(half h) = C[m0 + j + 8h][n0+l]
// ---------------------------------------------------------------------------
template <int RELU_A>
__global__ __launch_bounds__(256) void gcn_gemm_wmma_f32(
        const float* __restrict__ A, const float* __restrict__ W,
        float* __restrict__ out, int nrows, int ntiles_m) {
    __shared__ float wlds[D * D];   // 64 KB: whole weight matrix

    // Stage W into LDS (coalesced float4 copies).
    {
        const float4* wv = (const float4*)W;
        float4*       lv = (float4*)wlds;
        for (int i = threadIdx.x; i < (D * D) / 4; i += 256) lv[i] = wv[i];
    }
    __syncthreads();

    const int wave = threadIdx.x >> 5;   // 0..7 -> column tile
    const int lane = threadIdx.x & 31;
    const int half = lane >> 4;          // 0/1
    const int l    = lane & 15;
    const int n0   = wave * 16;

    for (int t = 0; t < MTILES; ++t) {
        const int mt = blockIdx.x * MTILES + t;   // uniform across block
        if (mt >= ntiles_m) break;                // uniform branch: EXEC intact
        const int m0 = mt * 16;

        int row = m0 + l;
        if (row >= nrows) row = nrows - 1;        // clamp load (store is guarded)
        const float* __restrict__ arow = A + (size_t)row * D;

        v8f c = {};
#pragma unroll
        for (int k0 = 0; k0 < D; k0 += 4) {
            v2f a = *(const v2f*)(arow + k0 + 2 * half);   // global_load_b64
            if (RELU_A) {                                   // fused ReLU (layer 2)
                a.x = fmaxf(a.x, 0.f);
                a.y = fmaxf(a.y, 0.f);
            }
            v2f b;
            b.x = wlds[(k0 + 2 * half) * D + n0 + l];       // ds_load_b32
            b.y = wlds[(k0 + 2 * half + 1) * D + n0 + l];
            // D = A(16x4) * B(4x16) + C, fp32 throughout
            c = __builtin_amdgcn_wmma_f32_16x16x4_f32(
                    /*neg_a=*/false, a, /*neg_b=*/false, b,
                    /*c_mod=*/(short)0, c, /*reuse_a=*/false, /*reuse_b=*/false);
        }

#pragma unroll
        for (int j = 0; j < 8; ++j) {
            const int r = m0 + j + 8 * half;
            if (r < nrows) out[(size_t)r * D + n0 + l] = c[j];
        }
    }
}

// ---------------------------------------------------------------------------
// COO SpMM scatter:  Y[row[e]] += val[e] * X[col[e]]   (D=128 channels)
// One wave32 per edge; lane moves one float4 (32 lanes x 16 B = 512 B row).
// Working set (X, Y ~51 MB each) fits the 192 MB L2, so the gathers hit L2;
// scatter uses non-returning global_atomic_add_f32 (STOREcnt path).
// ---------------------------------------------------------------------------
__global__ __launch_bounds__(256) void gcn_spmm_scatter(
        const float* __restrict__ vals, const int* __restrict__ rows,
        const int* __restrict__ cols, const float* __restrict__ X,
        float* __restrict__ Y, int nedges) {
    const int e    = (int)(((unsigned)blockIdx.x * blockDim.x + threadIdx.x) >> 5);
    const int lane = threadIdx.x & 31;
    if (e >= nedges) return;

    const float v = vals[e];     // same addr across wave -> broadcast from L2
    const int   r = rows[e];
    const int   c = cols[e];

    const float4 d = ((const float4*)(X + (size_t)c * D))[lane];
    float* dst = Y + (size_t)r * D + lane * 4;
    unsafeAtomicAdd(dst + 0, v * d.x);
    unsafeAtomicAdd(dst + 1, v * d.y);
    unsafeAtomicAdd(dst + 2, v * d.z);
    unsafeAtomicAdd(dst + 3, v * d.w);
}

// ---------------------------------------------------------------------------
// Pipeline:
//   T  = X @ W0                      (WMMA fp32)
//   Y1 = 0 ; Y1 += A-scatter(T)      (atomics)
//   T  = relu(Y1) @ W1               (WMMA fp32, ReLU fused into A load)
//   out= 0 ; out += A-scatter(T)     (atomics)
//   out= relu(out)
// Workspace: two nrows*128 fp32 buffers (T reused across layers, Y1).
// ---------------------------------------------------------------------------
extern "C" void kernel_launch(void* const* d_in, const int* in_sizes, int n_in,
                              void* d_out, int out_size, void* d_ws, size_t ws_size,
                              hipStream_t stream) {
    const float* X  = (const float*)d_in[0];   // node_feats [N,128]
    const float* W0 = (const float*)d_in[1];   // [128,128]
    const float* W1 = (const float*)d_in[2];   // [128,128]
    const float* ev = (const float*)d_in[3];   // edge_vals [E]
    const int*   er = (const int*)d_in[4];     // edge_row  [E]
    const int*   ec = (const int*)d_in[5];     // edge_col  [E]
    float* out = (float*)d_out;

    const int nrows  = in_sizes[0] / D;
    const int nedges = in_sizes[3];

    float* T  = (float*)d_ws;                      // nrows*128 floats
    float* Y1 = T + (size_t)nrows * D;             // nrows*128 floats

    const int  ntm     = (nrows + 15) / 16;
    const int  gblocks = (ntm + MTILES - 1) / MTILES;
    const long n4      = (long)nrows * D / 4;
    const int  zblocks = (int)((n4 + 255) / 256);
    const int  sblocks = (nedges + 7) / 8;         // 8 waves (edges) per block

    // ---- layer 1 ----
    gcn_gemm_wmma_f32<0><<<gblocks, 256, 0, stream>>>(X, W0, T, nrows, ntm);
    gcn_zero_kernel<<<zblocks, 256, 0, stream>>>((float4*)Y1, n4);
    gcn_spmm_scatter<<<sblocks, 256, 0, stream>>>(ev, er, ec, T, Y1, nedges);

    // ---- layer 2 (ReLU fused into GEMM A-operand load) ----
    gcn_gemm_wmma_f32<1><<<gblocks, 256, 0, stream>>>(Y1, W1, T, nrows, ntm);
    gcn_zero_kernel<<<zblocks, 256, 0, stream>>>((float4*)out, n4);
    gcn_spmm_scatter<<<sblocks, 256, 0, stream>>>(ev, er, ec, T, out, nedges);
    gcn_relu_kernel<<<zblocks, 256, 0, stream>>>((float4*)out, n4);
}